// ConditionalSplineSQ2D_37572373905830
// MI455X (gfx1250) — compile-verified
//
#include <hip/hip_runtime.h>

typedef __attribute__((ext_vector_type(2))) float v2f;
typedef __attribute__((ext_vector_type(8))) float v8f;
typedef __attribute__((ext_vector_type(4))) unsigned int u32x4;
typedef __attribute__((ext_vector_type(8))) int i32x8;
typedef __attribute__((ext_vector_type(4))) int i32x4;

#define KC 31          // K-1 intervals
#define NP 136         // triu pairs of 16
#define BM_TOTAL 512   // B*M
#define POLY_SLICE (2 * KC * 4)   // 248 floats per (b,m)

// -------------------------------------------------------------------------
// Kernel A: Hermite spline coefficients via V_WMMA_F32_16X16X4_F32.
// One wave handles 16 (b,m) rows for one (s,k) interval:
//   D[m, n] = sum_K A[m,K] * B[K,n],  A = [y0 y1 d0 d1] per row,
//   B[:,n] = n-th row of the 4x4 Hermite basis matrix (cols 4..15 zero).
// -------------------------------------------------------------------------
__global__ void poly_wmma_kernel(const float* __restrict__ params,
                                 const float* __restrict__ knots,
                                 float* __restrict__ out_poly) {
    const int lane = threadIdx.x & 31;
    const int widx = threadIdx.x >> 5;
    const int wg   = blockIdx.x * 8 + widx;          // 0 .. 1983
    const int bmTile = wg / 62;                      // 32 tiles of 16 (b,m) rows
    const int rem    = wg % 62;
    const int s = rem / KC;                          // 0..1 (x / y spline)
    const int k = rem % KC;                          // 0..30 interval

    // ---- A matrix: lanes 0-15 hold K=0,1 (y0,y1); lanes 16-31 hold K=2,3 (d0,d1)
    const int mrow = lane & 15;
    const int bm   = bmTile * 16 + mrow;
    const int t    = (lane < 16) ? 0 : 1;            // value vs derivative slot
    // params flat layout [B,M,2,K,2]: ((bm*2+s)*32+kk)*2 + t
    const int i0 = ((bm * 2 + s) * 32 + k) * 2 + t;
    const int i1 = ((bm * 2 + s) * 32 + (k + 1)) * 2 + t;
    v2f a;
    a.x = params[i0];
    a.y = params[i1];

    // ---- B matrix from Hermite basis, h = knots[k+1,s] - knots[k,s]
    const float h    = knots[(k + 1) * 2 + s] - knots[k * 2 + s];
    const float inv  = 1.0f / h;
    const float inv2 = inv * inv;
    const float inv3 = inv2 * inv;

    const int n = lane & 15;                         // output coefficient index
    float bx_lo = (n == 0) ? 1.0f
                : (n == 2) ? -3.0f * inv2
                : (n == 3) ?  2.0f * inv3 : 0.0f;    // M[n][0]  (y0 weight)
    float by_lo = (n == 2) ?  3.0f * inv2
                : (n == 3) ? -2.0f * inv3 : 0.0f;    // M[n][1]  (y1 weight)
    float bx_hi = (n == 1) ? 1.0f
                : (n == 2) ? -2.0f * inv
                : (n == 3) ?  inv2 : 0.0f;           // M[n][2]  (d0 weight)
    float by_hi = (n == 2) ? -inv
                : (n == 3) ?  inv2 : 0.0f;           // M[n][3]  (d1 weight)
    v2f bv;
    bv.x = (lane < 16) ? bx_lo : bx_hi;
    bv.y = (lane < 16) ? by_lo : by_hi;

    v8f c = {};
    v8f d = __builtin_amdgcn_wmma_f32_16x16x4_f32(
        /*neg_a=*/false, a, /*neg_b=*/false, bv,
        /*c_mod=*/(short)0, c, /*reuse_a=*/false, /*reuse_b=*/false);

    // ---- Scatter D: VGPR r holds row M=r (lanes 0-15) / M=r+8 (lanes 16-31),
    //      column N = lane&15; only N<4 are real coefficients c0..c3.
    const int cIdx = lane & 15;
    if (cIdx < 4) {
#pragma unroll
        for (int r = 0; r < 8; ++r) {
            const int mlocal = (lane < 16) ? r : (r + 8);
            const int bmrow  = bmTile * 16 + mlocal;
            // poly flat layout [B,M,2,KC,4]
            out_poly[((bmrow * 2 + s) * KC + k) * 4 + cIdx] = d[r];
        }
    }
}

// -------------------------------------------------------------------------
// Kernel B: Z[bm,i,j] = sum_p coeffs[i,j,p] * X[i,p] * Y[j,p]
// where X[i,p] = cx[i,II>>2]*cx[i,JJ>>2], Y[j,p] = cy[j,II&3]*cy[j,JJ&3].
// One block per (b,m). The block's poly slice is staged into LDS by the
// Tensor Data Mover (tensor_load_to_lds + s_wait_tensorcnt), then X/Y factor
// tables are built in LDS and coeffs streamed from L2 as float4.
// -------------------------------------------------------------------------
__global__ void z_kernel(const float* __restrict__ coeffs,
                         const float* __restrict__ poly,
                         float* __restrict__ out_z) {
    __shared__ __align__(16) float cpoly[POLY_SLICE];   // [s][i][a], 248 floats
    __shared__ unsigned char iu[NP], ju[NP];            // triu_indices(16)
    __shared__ __align__(16) float X[KC * NP];          // 31x136
    __shared__ __align__(16) float Y[KC * NP];

    const int bm  = blockIdx.x;                         // 0..511
    const int tid = threadIdx.x;

    // Warm L2 with the coeffs rows this block will stream (global_prefetch_b8).
    for (int e = tid; e < KC * KC; e += 256)
        __builtin_prefetch(coeffs + (size_t)e * NP, 0, 1);

    // ---- TDM: stage this (b,m)'s poly slice [2][31][4] (992 B) into LDS.
    // Wave 0 issues the descriptor (uniform branch: threads 0..31 = wave 0).
    if (tid < 32) {
        const unsigned int lds_addr =
            (unsigned int)(unsigned long long)(const void*)cpoly; // LDS byte offset
        const unsigned long long ga =
            (unsigned long long)(const void*)(poly + (size_t)bm * POLY_SLICE);

        u32x4 g0;
        g0.x = 1u;                                        // count=1 (valid), user D#
        g0.y = lds_addr;                                  // lds_addr (bytes)
        g0.z = (unsigned int)ga;                          // global_addr[31:0]
        g0.w = (unsigned int)((ga >> 32) & 0x01FFFFFFull) // global_addr[56:32]
             | (2u << 30);                                // type = 2 ("image")

        i32x8 g1;
        g1[0] = (int)(2u << 16);                 // workgroup_mask=0, data_size=2 (4B)
        g1[1] = (int)(248u << 16);               // abar_addr=0 | tensor_dim0[15:0]
        g1[2] = (int)((248u >> 16) | (1u << 16)); // tensor_dim0[31:16] | tensor_dim1[15:0]=1
        g1[3] = (int)(248u << 16);               // tensor_dim1[31:16]=0 | tile_dim0=248
        g1[4] = (int)1;                          // tile_dim1=1 | tile_dim2=0
        g1[5] = (int)248;                        // tensor_dim0_stride[31:0]
        g1[6] = 0;                               // stride hi | tensor_dim1_stride lo
        g1[7] = 0;                               // tensor_dim1_stride hi

        i32x4 gz  = {0, 0, 0, 0};                // groups 2/3 unused (<=2D tensor)
        i32x8 gz8 = {0, 0, 0, 0, 0, 0, 0, 0};    // trailing group (clang-23 form)
        __builtin_amdgcn_tensor_load_to_lds(g0, g1, gz, gz, gz8, 0);
        __builtin_amdgcn_s_wait_tensorcnt(0);
    }

    // Build triu index tables.
    if (tid < NP) {
        int u = 0, cum = 0;
        while (cum + (16 - u) <= tid) { cum += 16 - u; ++u; }
        const int v = u + (tid - cum);
        iu[tid] = (unsigned char)u;
        ju[tid] = (unsigned char)v;
    }
    __syncthreads();

    // Factorized pair products.
    for (int idx = tid; idx < KC * NP; idx += 256) {
        const int i = idx / NP;
        const int p = idx - i * NP;
        const int u = iu[p], v = ju[p];
        const float* cx = &cpoly[i * 4];                // s = 0
        const float* cy = &cpoly[(KC + i) * 4];         // s = 1
        X[idx] = cx[u >> 2] * cx[v >> 2];
        Y[idx] = cy[u & 3] * cy[v & 3];
    }
    __syncthreads();

    // Weighted contraction over p (136 = 34 float4 groups, all 16B-aligned).
    for (int e = tid; e < KC * KC; e += 256) {
        const int i = e / KC;
        const int j = e - i * KC;
        const float4* cp = (const float4*)(coeffs + (size_t)e * NP);
        const float4* xp = (const float4*)(X + i * NP);
        const float4* yp = (const float4*)(Y + j * NP);
        float acc = 0.0f;
#pragma unroll 4
        for (int q = 0; q < NP / 4; ++q) {
            const float4 cq = cp[q], xq = xp[q], yq = yp[q];
            acc = fmaf(cq.x * xq.x, yq.x, acc);
            acc = fmaf(cq.y * xq.y, yq.y, acc);
            acc = fmaf(cq.z * xq.z, yq.z, acc);
            acc = fmaf(cq.w * xq.w, yq.w, acc);
        }
        out_z[bm * (KC * KC) + e] = acc;
    }
}

extern "C" void kernel_launch(void* const* d_in, const int* in_sizes, int n_in,
                              void* d_out, int out_size, void* d_ws, size_t ws_size,
                              hipStream_t stream) {
    (void)in_sizes; (void)n_in; (void)d_ws; (void)ws_size; (void)out_size;
    const float* params = (const float*)d_in[0];   // [32,16,2,32,2]
    const float* knots  = (const float*)d_in[1];   // [32,2]
    const float* coeffs = (const float*)d_in[2];   // [31,31,136]

    float* out_poly = (float*)d_out;                       // 512*2*31*4 floats
    float* out_z    = (float*)d_out + BM_TOTAL * POLY_SLICE;

    // 32 bm-tiles * 2 splines * 31 intervals = 1984 wave-tasks, 8 waves/block.
    poly_wmma_kernel<<<248, 256, 0, stream>>>(params, knots, out_poly);
    z_kernel<<<BM_TOTAL, 256, 0, stream>>>(coeffs, out_poly, out_z);
}